// HyperSAGE_pyg_15255723835409
// MI455X (gfx1250) — compile-verified
//
#include <hip/hip_runtime.h>
#include <hip/hip_bf16.h>
#include <math.h>

typedef __attribute__((ext_vector_type(2))) float v2f;
typedef __attribute__((ext_vector_type(8))) float v8f;

#ifndef CLAMP_MIN_F
#define CLAMP_MIN_F 1e-7f
#define CLAMP_MAX_F 10.0f
#endif

__device__ __forceinline__ float clampHM(float v) {
    v = fmaxf(v, CLAMP_MIN_F);
    v = fminf(v, CLAMP_MAX_F);
    return v;
}

// ---------------------------------------------------------------------------
// Kernel 1: S = clip(x)  (scatter accumulator base; x itself stays read-only
// so the edge kernel can gather clip(x)^2 race-free while atomics hit S)
// ---------------------------------------------------------------------------
__global__ void clip_copy_kernel(const float* __restrict__ x,
                                 float* __restrict__ S, int n) {
    int i = blockIdx.x * blockDim.x + threadIdx.x;
    if (i < n) S[i] = clampHM(x[i]);
}

// ---------------------------------------------------------------------------
// Kernel 2: per-(edge,node) power-mean contributions, scatter-add into S.
//   D = feature width, EPB = edges per 128-thread block.
//   Each thread owns one feature column of one edge: gathers the 32 node
//   values (clipped^2), forms the edge sum, then atomically adds
//   sqrt((sum - self)/31) back to each member node's row in S.
//   k = 32 hardcoded (matches setup), power = 2 -> inv_p = 0.5 -> sqrtf.
// ---------------------------------------------------------------------------
template<int D, bool CLIP_ON_LOAD>
__global__ void edge_agg_kernel(const float* __restrict__ H,     // [N, D] gather source (read-only)
                                const int*   __restrict__ edges, // [E, 32]
                                float*       __restrict__ S,     // [N, D] accumulator
                                int E) {
    constexpr int EPB = 128 / D;
    __shared__ int idx[EPB][32];

    const int t  = threadIdx.x % D;   // feature column
    const int es = threadIdx.x / D;   // edge slot in block
    const int e  = blockIdx.x * EPB + es;

    for (int i = threadIdx.x; i < EPB * 32; i += blockDim.x) {
        int ee = blockIdx.x * EPB + (i >> 5);
        idx[i >> 5][i & 31] = (ee < E) ? edges[ee * 32 + (i & 31)] : 0;
    }
    __syncthreads();
    if (e >= E) return;

    float hp[32];
    float s = 0.f;
#pragma unroll
    for (int n = 0; n < 32; ++n) {
        float v = H[(long)idx[es][n] * D + t];
        if (CLIP_ON_LOAD) v = clampHM(v);
        float q = v * v;
        hp[n] = q;
        s += q;
    }
    const float invk1 = 1.0f / 31.0f;
#pragma unroll
    for (int n = 0; n < 32; ++n) {
        float c = sqrtf((s - hp[n]) * invk1);
        atomicAdd(&S[(long)idx[es][n] * D + t], c);
    }
}

// ---------------------------------------------------------------------------
// Kernel 3: fused row-normalize + GEMM + bias (+ optional relu-clip) via
// V_WMMA_F32_16X16X4_F32.  One wave = one 16-row tile of S [N,K].
// Normalization commutes with the GEMM (row scale), applied in the epilogue.
//   A fragment (16x4 f32, 2 VGPR): lane l (half h=l/16, r=l%16) holds
//     S[r][kb+2h], S[r][kb+2h+1]
//   B fragment (4x16 f32, 2 VGPR): lane l holds W[kb+2h][n0+r], W[kb+2h+1][n0+r]
//   C (16x16 f32, 8 VGPR): c[i] is element (M = i + 8h, N = r)
// ---------------------------------------------------------------------------
template<int K, int NOUT, bool RELU_CLIP>
__global__ void norm_gemm_kernel(const float* __restrict__ S,    // [N, K]
                                 const float* __restrict__ W,    // [K, NOUT] row-major
                                 const float* __restrict__ bias, // [NOUT]
                                 float*       __restrict__ out,  // [N, NOUT]
                                 float*       __restrict__ out2, // dup store (or null)
                                 int N) {
    const int lane = threadIdx.x & 31;
    const int wave = threadIdx.x >> 5;
    const int half = lane >> 4;
    const int lr   = lane & 15;

    const int tile = blockIdx.x * (blockDim.x >> 5) + wave;
    const int r0   = tile * 16;
    if (r0 >= N) return;                       // uniform per wave: EXEC stays full

    int rowA = r0 + lr;
    if (rowA > N - 1) rowA = N - 1;            // tail: clamp loads, guard stores

    // row sums: half 0 sums even k, half 1 odd k, combine across halves
    float psum = 0.f;
    for (int k = half; k < K; k += 2) psum += S[(long)rowA * K + k];
    psum += __shfl_xor(psum, 16, 32);
    const float rinv = (psum != 0.f) ? 1.0f / psum : 0.0f;

    // broadcast r_inv into C-layout row order
    float ri[8];
#pragma unroll
    for (int i = 0; i < 8; ++i) ri[i] = __shfl(rinv, i + 8 * half, 32);

    constexpr int NT = (NOUT + 15) / 16;
#pragma unroll
    for (int ct = 0; ct < NT; ++ct) {
        const int n0  = ct * 16;
        const int col = n0 + lr;
        const bool cv = (col < NOUT);
        v8f c = {};
        for (int j = 0; j < K / 4; ++j) {
            const int kb = 4 * j + 2 * half;
            v2f a, b;
            a.x = S[(long)rowA * K + kb];
            a.y = S[(long)rowA * K + kb + 1];
            b.x = cv ? W[kb * NOUT + col] : 0.f;
            b.y = cv ? W[(kb + 1) * NOUT + col] : 0.f;
            c = __builtin_amdgcn_wmma_f32_16x16x4_f32(
                    /*neg_a=*/false, a, /*neg_b=*/false, b,
                    /*c_mod=*/(short)0, c, /*reuse_a=*/false, /*reuse_b=*/false);
        }
        const float bv = cv ? bias[col] : 0.f;
#pragma unroll
        for (int i = 0; i < 8; ++i) {
            const int  m   = i + 8 * half;
            const long row = (long)r0 + m;
            float v = c[i] * ri[i] + bv;
            if (RELU_CLIP) v = clampHM(v);     // clip(relu(v),1e-7,10) == clip(v,...)
            if (row < N && cv) {
                out[row * NOUT + col] = v;
                if (out2) out2[row * NOUT + col] = v;
            }
        }
    }
}

// ---------------------------------------------------------------------------
// Launch: layer1 (d=128 -> 16, relu) then layer2 (16 -> 40).
// Workspace layout: [S1: N*128 f32][Hc2: N*16 f32][S2: N*16 f32]  (~64 MB)
// ---------------------------------------------------------------------------
extern "C" void kernel_launch(void* const* d_in, const int* in_sizes, int n_in,
                              void* d_out, int out_size, void* d_ws, size_t ws_size,
                              hipStream_t stream) {
    const float* x     = (const float*)d_in[0];   // [N,128]
    const int*   edges = (const int*)  d_in[1];   // [E,32]
    const float* W1    = (const float*)d_in[2];   // [128,16]
    const float* b1    = (const float*)d_in[3];   // [16]
    const float* W2    = (const float*)d_in[4];   // [16,40]
    const float* b2    = (const float*)d_in[5];   // [40]
    float*       out   = (float*)d_out;           // [N,40]

    const int N = in_sizes[0] / 128;
    const int E = in_sizes[1] / 32;

    float* S1  = (float*)d_ws;           // [N,128] layer-1 accumulator
    float* Hc2 = S1 + (long)N * 128;     // [N,16]  clipped layer-1 output (gather src)
    float* S2  = Hc2 + (long)N * 16;     // [N,16]  layer-2 accumulator

    // ---- layer 1 ----
    {
        int n = N * 128;
        clip_copy_kernel<<<(n + 255) / 256, 256, 0, stream>>>(x, S1, n);
        edge_agg_kernel<128, true><<<E, 128, 0, stream>>>(x, edges, S1, E);
        int tiles  = (N + 15) / 16;
        int blocks = (tiles + 3) / 4;    // 4 waves/block
        norm_gemm_kernel<128, 16, true><<<blocks, 128, 0, stream>>>(
            S1, W1, b1, Hc2, /*dup->*/S2, N);
    }
    // ---- layer 2 ----
    {
        edge_agg_kernel<16, false><<<(E + 7) / 8, 128, 0, stream>>>(Hc2, edges, S2, E);
        int tiles  = (N + 15) / 16;
        int blocks = (tiles + 3) / 4;
        norm_gemm_kernel<16, 40, false><<<blocks, 128, 0, stream>>>(
            S2, W2, b2, out, nullptr, N);
    }
}